// InsectAwareProtoPool_1700807049514
// MI455X (gfx1250) — compile-verified
//
#include <hip/hip_runtime.h>

// ---------------------------------------------------------------------------
// InsectAwareProtoPool on MI455X (gfx1250, wave32)
//
// Two kernels:
//  1) proto_mean_wmma: mean_table[c,s,:] = sum_p class_protos[c,s,p,:] / max(cnt,1)
//     done as ones(1x8) x M(8x768) via two chained V_WMMA_F32_16X16X4_F32 per
//     16-column block (A = all-ones makes the result invariant to the exact
//     K/M VGPR permutation; only N=lane mapping of B and D matters).
//     Also computes shared_mean[s,:] (tiny) in one extra block.
//  2) gather_axpy: out = feat + 0.5*(mean_table[cid*4+st] + shared_mean[st])
//     one wave per row, 6 x b128 per lane, NT hints on the 384MB stream so the
//     12MB mean table stays resident in the 192MB L2 for the random gathers.
// ---------------------------------------------------------------------------

typedef float v2f __attribute__((ext_vector_type(2)));
typedef float v4f __attribute__((ext_vector_type(4)));
typedef float v8f __attribute__((ext_vector_type(8)));

#define DIM         768
#define NUM_CLASSES 1000
#define MAX_STAGES  4
#define MAX_PROTO   8
#define SHARED_P    16
#define CS_TOTAL    (NUM_CLASSES * MAX_STAGES)   // 4000
#define NROWS       65536

// ---------------------------------------------------------------------------
// Kernel 1: per-(class,stage) prototype mean via WMMA + shared-proto mean.
// Grid: 1001 blocks x 256 threads. Blocks 0..999 = 8000 waves, 2 waves per
// (c,s) pair (each wave covers 24 of the 48 16-column blocks). Block 1000
// computes shared_mean.
// ---------------------------------------------------------------------------
__global__ __launch_bounds__(256) void proto_mean_wmma(
    const float* __restrict__ class_protos,   // [1000,4,8,768]
    const int*   __restrict__ proto_counts,   // [1000,4]
    const float* __restrict__ shared_protos,  // [4,16,768]
    float* __restrict__ mean_table,           // [4000,768]
    float* __restrict__ shared_mean)          // [4,768]
{
    if (blockIdx.x == 1000) {
        // shared_mean[s,d] = (1/16) * sum_p shared_protos[s,p,d]  (49K floats)
        int tid = threadIdx.x;
        for (int d = tid; d < DIM; d += 256) {
            for (int s = 0; s < MAX_STAGES; ++s) {
                const float* p = shared_protos + (size_t)(s * SHARED_P) * DIM + d;
                float acc = 0.0f;
#pragma unroll
                for (int k = 0; k < SHARED_P; ++k) acc += p[(size_t)k * DIM];
                shared_mean[s * DIM + d] = acc * (1.0f / (float)SHARED_P);
            }
        }
        return;
    }

    const int w       = (blockIdx.x << 3) + (threadIdx.x >> 5); // wave id 0..7999
    const int cs      = w >> 1;                                 // (c,s) pair 0..3999
    const int colhalf = w & 1;                                  // which 384 columns
    const int lane    = threadIdx.x & 31;
    const int half    = lane >> 4;                              // lane group 0/1
    const int nl      = lane & 15;                              // column within block

    const int   cnt = proto_counts[cs];
    const float inv = 1.0f / (float)(cnt > 0 ? cnt : 1);

    // B-matrix lane base: VGPR0 holds K = 2*half, VGPR1 holds K = 2*half+1
    // (mirrors the documented 16x4 A layout; result is K-permutation-invariant
    //  anyway since A is all ones).
    const float* base = class_protos + (size_t)cs * (MAX_PROTO * DIM)
                                     + (size_t)(2 * half) * DIM + nl;
    float* outrow = mean_table + (size_t)cs * DIM;

    v2f ones; ones.x = 1.0f; ones.y = 1.0f;   // A = ones(16x4), layout-invariant

#pragma unroll 4
    for (int nb = 0; nb < 24; ++nb) {
        const int n0 = (colhalf * 24 + nb) << 4;     // column block start

        // Group 0: protos 0..3 ; Group 1: protos 4..7
        v2f bA, bB;
        bA.x = base[n0];                 // proto 2*half
        bA.y = base[n0 + DIM];           // proto 2*half+1
        bB.x = base[n0 + 4 * DIM];       // proto 4+2*half
        bB.y = base[n0 + 5 * DIM];       // proto 5+2*half

        v8f acc = {};
        // D = ones x B + C : every output row = column sum over the 4 K-rows.
        acc = __builtin_amdgcn_wmma_f32_16x16x4_f32(
                  false, ones, false, bA, (short)0, acc, false, false);
        acc = __builtin_amdgcn_wmma_f32_16x16x4_f32(
                  false, ones, false, bB, (short)0, acc, false, false);

        // D row M=0 lives in VGPR0 (acc[0]) lanes 0..15, N = lane.
        if (lane < 16) outrow[n0 + lane] = acc[0] * inv;
    }
}

// ---------------------------------------------------------------------------
// Kernel 2: out = feat + 0.5*(mean_table[cid*4+st] + shared_mean[st])
// One wave per row; 192 float4 per row -> 6 b128 per lane.
// NT load/store on the streamed features/out; RT (default) on table gathers.
// ---------------------------------------------------------------------------
__global__ __launch_bounds__(256) void gather_axpy(
    const float* __restrict__ features,     // [65536,768]
    const int*   __restrict__ class_ids,    // [65536]
    const int*   __restrict__ stages,       // [65536]
    const float* __restrict__ mean_table,   // [4000,768]
    const float* __restrict__ shared_mean,  // [4,768]
    float* __restrict__ out)                // [65536,768]
{
    const int row  = (blockIdx.x << 3) + (threadIdx.x >> 5);  // wave-uniform
    const int lane = threadIdx.x & 31;

    // Force scalar (SADDR-form) addressing for the gathered rows.
    const int cid = __builtin_amdgcn_readfirstlane(class_ids[row]);
    const int st  = __builtin_amdgcn_readfirstlane(stages[row]);

    const v4f* frow = (const v4f*)(features   + (size_t)row * DIM);
    const v4f* trow = (const v4f*)(mean_table + (size_t)(cid * MAX_STAGES + st) * DIM);
    const v4f* srow = (const v4f*)(shared_mean + (size_t)st * DIM);
    v4f*       orow = (v4f*)(out + (size_t)row * DIM);

#pragma unroll
    for (int i = 0; i < 6; ++i) {
        const int idx = lane + (i << 5);
        v4f f = __builtin_nontemporal_load(frow + idx);  // streamed, don't cache
        v4f t = trow[idx];                               // L2-resident (12MB table)
        v4f s = srow[idx];                               // L2/L0-resident (12KB)
        v4f r = f + 0.5f * (t + s);
        __builtin_nontemporal_store(r, orow + idx);      // streamed out
    }
}

// ---------------------------------------------------------------------------
extern "C" void kernel_launch(void* const* d_in, const int* in_sizes, int n_in,
                              void* d_out, int out_size, void* d_ws, size_t ws_size,
                              hipStream_t stream) {
    const float* features      = (const float*)d_in[0];  // [65536,768]
    const int*   class_ids     = (const int*)  d_in[1];  // [65536]
    const int*   stages        = (const int*)  d_in[2];  // [65536]
    const float* class_protos  = (const float*)d_in[3];  // [1000,4,8,768]
    const int*   proto_counts  = (const int*)  d_in[4];  // [1000,4]
    const float* shared_protos = (const float*)d_in[5];  // [4,16,768]
    float*       out           = (float*)d_out;

    float* mean_table  = (float*)d_ws;                    // 4000*768 floats
    float* shared_mean = mean_table + CS_TOTAL * DIM;     // + 4*768 floats

    // Blocks 0..999: 8000 waves = 2 waves per (c,s). Block 1000: shared mean.
    proto_mean_wmma<<<1001, 256, 0, stream>>>(
        class_protos, proto_counts, shared_protos, mean_table, shared_mean);

    // 65536 rows / 8 waves per block = 8192 blocks.
    gather_axpy<<<NROWS / 8, 256, 0, stream>>>(
        features, class_ids, stages, mean_table, shared_mean, out);
}